// Graph_PaiNN_55078660604618
// MI455X (gfx1250) — compile-verified
//
#include <hip/hip_runtime.h>
#include <math.h>

#define N_ATOMS 10000
#define N_PAIRS 160000
#define FDIM    128
#define RDIM    64
#define NBLK    5
#define EPS_C   1e-8f
#define CHUNK   16384   // pair-chunk so Wij (25MB) stays L2-resident between producer/consumer

typedef __attribute__((ext_vector_type(2))) float v2f;
typedef __attribute__((ext_vector_type(8))) float v8f;

// ---------------------------------------------------------------------------
// LDS-staged fp32 WMMA GEMM:  C[M,Nc] = epilogue(A[M,K] @ W[K,Nc])
//   epilogue: +bias[n] (optional), SiLU (act==1), * rscale[m] (optional)
//
// Block = 256 threads (8 waves), computes a 16 x 128 C tile:
//   wave w owns the 16x16 sub-tile at columns n0 + 16*w.
// K is walked in 32-wide panels staged in LDS:
//   As[16][34]  (pad 34: even -> float2-aligned; 34*r mod 64 distinct -> no
//                bank conflicts on the 16-lane column reads)
//   Bs[32][132] (pad 132: rows stay 16B aligned for float4 stores)
// Inner loop: 8 x v_wmma_f32_16x16x4_f32 per K-panel, fed from ds_load.
// Requirements (all satisfied by every GEMM here): M%16==0, Nc%128==0, K%32==0.
// Grid sized exactly -> no early-out -> EXEC all-ones for WMMA.
// ---------------------------------------------------------------------------
__global__ void __launch_bounds__(256)
wmma_gemm_f32(const float* __restrict__ A, int lda,
              const float* __restrict__ W, int ldw,
              const float* __restrict__ bias,
              const float* __restrict__ rscale,
              float* __restrict__ C, int ldc,
              int M, int K, int Nc, int act)
{
    __shared__ float As[16][34];
    __shared__ float Bs[32][132];

    const int ntb = Nc >> 7;                 // N super-tiles of 128
    const int tm  = blockIdx.x / ntb;
    const int tn  = blockIdx.x % ntb;
    const int m0  = tm << 4;
    const int n0  = tn << 7;

    const int tid  = threadIdx.x;
    const int wave = tid >> 5;
    const int lane = tid & 31;
    const int grp  = lane >> 4;              // 0: K=k,k+1 ; 1: K=k+2,k+3
    const int lrow = lane & 15;
    const int ncol = (wave << 4) + lrow;     // this lane's column inside the 128-wide tile

    // A-tile loader: thread t -> row t>>4, cols 2*(t&15), 2*(t&15)+1
    const int ar = tid >> 4;
    const int ac = (tid & 15) << 1;
    const float* Abase = A + (size_t)(m0 + ar) * lda + ac;
    // B-tile loader: thread t -> rows (t>>5)+8*i, cols 4*(t&31)
    const int br = tid >> 5;
    const int bc = (tid & 31) << 2;
    const float* Bbase = W + (size_t)br * ldw + n0 + bc;

    v8f acc = {0.f, 0.f, 0.f, 0.f, 0.f, 0.f, 0.f, 0.f};

    for (int k0 = 0; k0 < K; k0 += 32) {
        // stage A 16x32 panel
        {
            const float* src = Abase + k0;
            As[ar][ac]     = src[0];
            As[ar][ac + 1] = src[1];
        }
        // stage B 32x128 panel (4 rows of 8 per thread-group)
        {
            const float* src = Bbase + (size_t)k0 * ldw;
#pragma unroll
            for (int i = 0; i < 4; ++i) {
                const float* s = src + (size_t)(i * 8) * ldw;
                float4 v = *(const float4*)s;
                Bs[br + 8 * i][bc]     = v.x;
                Bs[br + 8 * i][bc + 1] = v.y;
                Bs[br + 8 * i][bc + 2] = v.z;
                Bs[br + 8 * i][bc + 3] = v.w;
            }
        }
        __syncthreads();

#pragma unroll
        for (int kk = 0; kk < 32; kk += 4) {
            const int ka = kk + (grp << 1);
            v2f a, b;
            a.x = As[lrow][ka];              // A[M=lrow, K=ka]
            a.y = As[lrow][ka + 1];          // A[M=lrow, K=ka+1]
            b.x = Bs[ka][ncol];              // B[K=ka,   N=ncol]
            b.y = Bs[ka + 1][ncol];          // B[K=ka+1, N=ncol]
            acc = __builtin_amdgcn_wmma_f32_16x16x4_f32(
                      false, a, false, b, (short)0, acc, false, false);
        }
        __syncthreads();
    }

    const int   n  = n0 + ncol;
    const float bn = bias ? bias[n] : 0.0f;
#pragma unroll
    for (int v = 0; v < 8; ++v) {
        const int m = m0 + v + (grp << 3);   // C row = v + 8*lanegroup
        float val = acc[v] + bn;
        if (act) val = val / (1.0f + __expf(-val));   // SiLU
        if (rscale) val *= rscale[m];
        C[(size_t)m * ldc + n] = val;
    }
}

// ---------------------------------------------------------------------------
// dirv[p] = vectors[p] / distances[p]
// ---------------------------------------------------------------------------
__global__ void dirv_kernel(const float* __restrict__ vec,
                            const float* __restrict__ dist,
                            float* __restrict__ dirv, int P)
{
    int p = blockIdx.x * blockDim.x + threadIdx.x;
    if (p >= P) return;
    float inv = 1.0f / dist[p];
    dirv[p * 3 + 0] = vec[p * 3 + 0] * inv;
    dirv[p * 3 + 1] = vec[p * 3 + 1] * inv;
    dirv[p * 3 + 2] = vec[p * 3 + 2] * inv;
}

// ---------------------------------------------------------------------------
// Edge gather -> filter -> scatter-add.  One block (128 threads) per pair;
// thread f handles feature f.  mu_in is a pre-update snapshot of mu.
// ---------------------------------------------------------------------------
__global__ void edge_kernel(const float* __restrict__ Wc,     // [cn, 3F] filter chunk
                            const float* __restrict__ x,      // [N, 3F]
                            const float* __restrict__ mu_in,  // [N, 3, F] snapshot
                            const float* __restrict__ dirv,   // [P, 3]
                            const int* __restrict__ idx_i,
                            const int* __restrict__ idx_j,
                            float* __restrict__ q,            // [N, F]  accumulated
                            float* __restrict__ mu,           // [N,3,F] accumulated
                            int chunk_start, int npairs)
{
    const int p = blockIdx.x;
    if (p >= npairs) return;
    const int f  = threadIdx.x;           // 0..127
    const int gp = chunk_start + p;
    const int i  = idx_i[gp];
    const int j  = idx_j[gp];

    const float* Wp = Wc + (size_t)p * (3 * FDIM);
    const float* xj = x  + (size_t)j * (3 * FDIM);

    const float dq    = xj[f]            * Wp[f];
    const float dmuR  = xj[FDIM + f]     * Wp[FDIM + f];
    const float dmumu = xj[2 * FDIM + f] * Wp[2 * FDIM + f];

    atomicAdd(&q[(size_t)i * FDIM + f], dq);

    const float d0 = dirv[gp * 3 + 0];
    const float d1 = dirv[gp * 3 + 1];
    const float d2 = dirv[gp * 3 + 2];
    const float dv[3] = {d0, d1, d2};
#pragma unroll
    for (int v = 0; v < 3; ++v) {
        const float muj = mu_in[((size_t)j * 3 + v) * FDIM + f];
        atomicAdd(&mu[((size_t)i * 3 + v) * FDIM + f], dmuR * dv[v] + dmumu * muj);
    }
}

// ---------------------------------------------------------------------------
// Mixing pre: mu_Vn = sqrt(sum_v mu_V^2 + eps); ctx = [q, mu_Vn]
// mu_mix layout: [N, 3, 2F] (rows of the 3N x 2F GEMM output)
// ---------------------------------------------------------------------------
__global__ void mix_pre_kernel(const float* __restrict__ q,
                               const float* __restrict__ mumix,
                               float* __restrict__ ctx)
{
    const int n = blockIdx.x;
    const int f = threadIdx.x;            // 0..127
    float s = EPS_C;
#pragma unroll
    for (int v = 0; v < 3; ++v) {
        float mv = mumix[((size_t)n * 3 + v) * (2 * FDIM) + f];   // mu_V
        s += mv * mv;
    }
    ctx[(size_t)n * (2 * FDIM) + f]        = q[(size_t)n * FDIM + f];
    ctx[(size_t)n * (2 * FDIM) + FDIM + f] = sqrtf(s);
}

// ---------------------------------------------------------------------------
// Mixing post: q += dq + dqmu * <mu_V, mu_W>;  mu += dmu * mu_W
// ---------------------------------------------------------------------------
__global__ void mix_post_kernel(const float* __restrict__ y,      // [N, 3F]
                                const float* __restrict__ mumix,  // [N, 3, 2F]
                                float* __restrict__ q,
                                float* __restrict__ mu)
{
    const int n = blockIdx.x;
    const int f = threadIdx.x;            // 0..127
    float muW[3];
    float s = 0.0f;
#pragma unroll
    for (int v = 0; v < 3; ++v) {
        const float mV = mumix[((size_t)n * 3 + v) * (2 * FDIM) + f];
        muW[v]         = mumix[((size_t)n * 3 + v) * (2 * FDIM) + FDIM + f];
        s += mV * muW[v];
    }
    const float dq   = y[(size_t)n * (3 * FDIM) + f];
    const float dmu  = y[(size_t)n * (3 * FDIM) + FDIM + f];
    const float dqmu = y[(size_t)n * (3 * FDIM) + 2 * FDIM + f];

    q[(size_t)n * FDIM + f] += dq + dqmu * s;
#pragma unroll
    for (int v = 0; v < 3; ++v)
        mu[((size_t)n * 3 + v) * FDIM + f] += dmu * muW[v];
}

// ---------------------------------------------------------------------------
static inline void launch_gemm(const float* A, int lda, const float* W, int ldw,
                               const float* bias, const float* rscale,
                               float* C, int ldc, int M, int K, int Nc, int act,
                               hipStream_t s)
{
    const int blocks = (M >> 4) * (Nc >> 7);         // exact: 16 x 128 tiles
    wmma_gemm_f32<<<blocks, 256, 0, s>>>(A, lda, W, ldw, bias, rscale, C, ldc,
                                         M, K, Nc, act);
}

extern "C" void kernel_launch(void* const* d_in, const int* in_sizes, int n_in,
                              void* d_out, int out_size, void* d_ws, size_t ws_size,
                              hipStream_t stream)
{
    const float* features  = (const float*)d_in[0];   // [N, F]
    const float* distances = (const float*)d_in[1];   // [P]
    const float* vectors   = (const float*)d_in[2];   // [P, 3]
    const float* cutoffs   = (const float*)d_in[3];   // [P]
    const float* rbfs      = (const float*)d_in[4];   // [P, R]
    const int*   idx_i     = (const int*)d_in[5];     // [P]
    const int*   idx_j     = (const int*)d_in[6];     // [P]
    const float* Wf        = (const float*)d_in[7];   // [R, B*3F]
    const float* bf        = (const float*)d_in[8];   // [B*3F]
    const float* iW1       = (const float*)d_in[9];   // [B, F, F]
    const float* ib1       = (const float*)d_in[10];  // [B, F]
    const float* iW2       = (const float*)d_in[11];  // [B, F, 3F]
    const float* ib2       = (const float*)d_in[12];  // [B, 3F]
    const float* mWmix     = (const float*)d_in[13];  // [B, F, 2F]
    const float* mW1       = (const float*)d_in[14];  // [B, 2F, F]
    const float* mb1       = (const float*)d_in[15];  // [B, F]
    const float* mW2       = (const float*)d_in[16];  // [B, F, 3F]
    const float* mb2       = (const float*)d_in[17];  // [B, 3F]

    float* q  = (float*)d_out;                         // [N, F]
    float* mu = q + (size_t)N_ATOMS * FDIM;            // [N, 3, F]

    // workspace carve-up (~125 MB of floats)
    float* ws    = (float*)d_ws;
    float* h     = ws;                                         // [N, F]
    float* x     = h     + (size_t)N_ATOMS * FDIM;             // [N, 3F]
    float* mu_in = x     + (size_t)N_ATOMS * 3 * FDIM;         // [N, 3, F]
    float* mumix = mu_in + (size_t)N_ATOMS * 3 * FDIM;         // [N, 3, 2F]
    float* ctx   = mumix + (size_t)N_ATOMS * 3 * 2 * FDIM;     // [N, 2F]
    float* tbuf  = ctx   + (size_t)N_ATOMS * 2 * FDIM;         // [N, F]
    float* ybuf  = tbuf  + (size_t)N_ATOMS * FDIM;             // [N, 3F]
    float* dirv  = ybuf  + (size_t)N_ATOMS * 3 * FDIM;         // [P, 3]
    float* Wc    = dirv  + (size_t)N_PAIRS * 3;                // [CHUNK, 3F]

    // init: q = features, mu = 0, dirv = vectors / distances
    hipMemcpyAsync(q, features, sizeof(float) * (size_t)N_ATOMS * FDIM,
                   hipMemcpyDeviceToDevice, stream);
    hipMemsetAsync(mu, 0, sizeof(float) * (size_t)N_ATOMS * 3 * FDIM, stream);
    dirv_kernel<<<(N_PAIRS + 255) / 256, 256, 0, stream>>>(vectors, distances,
                                                           dirv, N_PAIRS);

    for (int b = 0; b < NBLK; ++b) {
        // ---- interaction atom-side GEMMs: x = silu(q@iW1+ib1)@iW2+ib2 ----
        launch_gemm(q, FDIM, iW1 + (size_t)b * FDIM * FDIM, FDIM,
                    ib1 + b * FDIM, nullptr, h, FDIM,
                    N_ATOMS, FDIM, FDIM, /*silu*/1, stream);
        launch_gemm(h, FDIM, iW2 + (size_t)b * FDIM * 3 * FDIM, 3 * FDIM,
                    ib2 + b * 3 * FDIM, nullptr, x, 3 * FDIM,
                    N_ATOMS, FDIM, 3 * FDIM, 0, stream);

        // snapshot mu (edge phase reads mu[idx_j] pre-update)
        hipMemcpyAsync(mu_in, mu, sizeof(float) * (size_t)N_ATOMS * 3 * FDIM,
                       hipMemcpyDeviceToDevice, stream);

        // ---- edge phase, chunked so Wij stays in L2 between kernels ----
        for (int cs = 0; cs < N_PAIRS; cs += CHUNK) {
            const int cn = (N_PAIRS - cs < CHUNK) ? (N_PAIRS - cs) : CHUNK;
            // Wij = (rbf_chunk @ Wf[:, b*3F:(b+1)*3F] + bf_slice) * cutoff
            launch_gemm(rbfs + (size_t)cs * RDIM, RDIM,
                        Wf + (size_t)b * 3 * FDIM, NBLK * 3 * FDIM,
                        bf + b * 3 * FDIM, cutoffs + cs,
                        Wc, 3 * FDIM, cn, RDIM, 3 * FDIM, 0, stream);
            edge_kernel<<<cn, FDIM, 0, stream>>>(Wc, x, mu_in, dirv,
                                                 idx_i, idx_j, q, mu, cs, cn);
        }

        // ---- mixing ----
        // mu_mix = mu @ mWmix  (rows = 3N)
        launch_gemm(mu, FDIM, mWmix + (size_t)b * FDIM * 2 * FDIM, 2 * FDIM,
                    nullptr, nullptr, mumix, 2 * FDIM,
                    3 * N_ATOMS, FDIM, 2 * FDIM, 0, stream);
        mix_pre_kernel<<<N_ATOMS, FDIM, 0, stream>>>(q, mumix, ctx);
        launch_gemm(ctx, 2 * FDIM, mW1 + (size_t)b * 2 * FDIM * FDIM, FDIM,
                    mb1 + b * FDIM, nullptr, tbuf, FDIM,
                    N_ATOMS, 2 * FDIM, FDIM, /*silu*/1, stream);
        launch_gemm(tbuf, FDIM, mW2 + (size_t)b * FDIM * 3 * FDIM, 3 * FDIM,
                    mb2 + b * 3 * FDIM, nullptr, ybuf, 3 * FDIM,
                    N_ATOMS, FDIM, 3 * FDIM, 0, stream);
        mix_post_kernel<<<N_ATOMS, FDIM, 0, stream>>>(ybuf, mumix, q, mu);
    }
}